// MyGATConv_89034672046801
// MI455X (gfx1250) — compile-verified
//
#include <hip/hip_runtime.h>
#include <hip/hip_bf16.h>

#define N_NODES 100000
#define N_EDGES 1600000
#define IN_DIM  256
#define H_HEADS 4
#define F_DIM   64
#define HF      256           // H_HEADS * F_DIM
#define NEG_SLOPE 0.05f

typedef __attribute__((ext_vector_type(2))) float v2f;
typedef __attribute__((ext_vector_type(8))) float v8f;

__device__ __forceinline__ float fadd_atomic(float* p, float v) {
    return __hip_atomic_fetch_add(p, v, __ATOMIC_RELAXED, __HIP_MEMORY_SCOPE_AGENT);
}

// monotonic uint encoding of f32 (order-preserving), so segment-max is a u32 atomicMax
__device__ __forceinline__ unsigned enc_f32(float f) {
    unsigned b = __float_as_uint(f);
    return (b & 0x80000000u) ? ~b : (b | 0x80000000u);
}
__device__ __forceinline__ float dec_f32(unsigned u) {
    unsigned b = (u & 0x80000000u) ? (u & 0x7FFFFFFFu) : ~u;
    return __uint_as_float(b);
}

// ---------------------------------------------------------------------------
// Fused dual GEMM + attention-dot epilogue.
//   feat_src = X @ W_fc^T                      -> d_ws
//   rst      = X @ W_res^T + bias              -> d_out (SpMM accumulates later)
//   el[n][h] = sum_f feat_src[n][h][f]*attn_l[h][f]   (nq quadrant == head h)
//   er[n][h] = sum_f feat_src[n][h][f]*attn_r[h][f]
// f32 WMMA 16x16x4. Block = 256 threads = 8 waves arranged 2(M) x 4(Nquad).
// Each wave: 16 rows x 64 cols, one A-frag feeds 8 WMMAs (4 fc + 4 res).
// ---------------------------------------------------------------------------
__global__ __launch_bounds__(256) void gat_fused_gemm(
    const float* __restrict__ X,      // [N,256]
    const float* __restrict__ Wfc,    // [256,256] row-major [out][in]
    const float* __restrict__ Wres,   // [256,256]
    const float* __restrict__ bias,   // [256]
    const float* __restrict__ attn_l, // [256] flat (h*64+f)
    const float* __restrict__ attn_r, // [256]
    float* __restrict__ feat_src,     // [N,256]
    float* __restrict__ rst,          // [N,256]
    float* __restrict__ el,           // [N,4]
    float* __restrict__ er)           // [N,4]
{
    const int lane = threadIdx.x & 31;
    const int wave = threadIdx.x >> 5;
    const int mi   = wave & 1;
    const int nq   = wave >> 1;       // == head h for this wave's 64 columns
    const int rowBase = blockIdx.x * 32 + mi * 16;
    const int colBase = nq * 64;

    // A 16x4 f32 frag: lane l holds A[l&15][2*(l>>4) + {0,1}]
    // B 4x16 f32 frag: lane l holds B[2*(l>>4) + {0,1}][l&15] = W[col][k..k+1]
    const int r  = lane & 15;
    const int kk = (lane >> 4) << 1;

    const float* xrow = X + (size_t)(rowBase + r) * IN_DIM + kk;
    const float* f0 = Wfc  + (size_t)(colBase +  0 + r) * IN_DIM + kk;
    const float* f1 = Wfc  + (size_t)(colBase + 16 + r) * IN_DIM + kk;
    const float* f2 = Wfc  + (size_t)(colBase + 32 + r) * IN_DIM + kk;
    const float* f3 = Wfc  + (size_t)(colBase + 48 + r) * IN_DIM + kk;
    const float* g0 = Wres + (size_t)(colBase +  0 + r) * IN_DIM + kk;
    const float* g1 = Wres + (size_t)(colBase + 16 + r) * IN_DIM + kk;
    const float* g2 = Wres + (size_t)(colBase + 32 + r) * IN_DIM + kk;
    const float* g3 = Wres + (size_t)(colBase + 48 + r) * IN_DIM + kk;

    v8f c0 = {}, c1 = {}, c2 = {}, c3 = {};   // W_fc accumulators
    v8f d0 = {}, d1 = {}, d2 = {}, d3 = {};   // W_res accumulators

    for (int k0 = 0; k0 < IN_DIM; k0 += 4) {
        v2f a  = *(const v2f*)(xrow + k0);
        v2f bf0 = *(const v2f*)(f0 + k0);
        v2f bf1 = *(const v2f*)(f1 + k0);
        v2f bf2 = *(const v2f*)(f2 + k0);
        v2f bf3 = *(const v2f*)(f3 + k0);
        v2f bg0 = *(const v2f*)(g0 + k0);
        v2f bg1 = *(const v2f*)(g1 + k0);
        v2f bg2 = *(const v2f*)(g2 + k0);
        v2f bg3 = *(const v2f*)(g3 + k0);
        c0 = __builtin_amdgcn_wmma_f32_16x16x4_f32(false, a, false, bf0, (short)0, c0, false, false);
        c1 = __builtin_amdgcn_wmma_f32_16x16x4_f32(false, a, false, bf1, (short)0, c1, false, false);
        c2 = __builtin_amdgcn_wmma_f32_16x16x4_f32(false, a, false, bf2, (short)0, c2, false, false);
        c3 = __builtin_amdgcn_wmma_f32_16x16x4_f32(false, a, false, bf3, (short)0, c3, false, false);
        d0 = __builtin_amdgcn_wmma_f32_16x16x4_f32(false, a, false, bg0, (short)0, d0, false, false);
        d1 = __builtin_amdgcn_wmma_f32_16x16x4_f32(false, a, false, bg1, (short)0, d1, false, false);
        d2 = __builtin_amdgcn_wmma_f32_16x16x4_f32(false, a, false, bg2, (short)0, d2, false, false);
        d3 = __builtin_amdgcn_wmma_f32_16x16x4_f32(false, a, false, bg3, (short)0, d3, false, false);
    }

    // C/D 16x16 f32 layout: lane l, VGPR v -> row = v + 8*(l>>4), col = l&15
    const int cr = (lane >> 4) << 3;
    const int cc = lane & 15;
    const float b0 = bias[colBase + cc];
    const float b1 = bias[colBase + 16 + cc];
    const float b2 = bias[colBase + 32 + cc];
    const float b3 = bias[colBase + 48 + cc];
    // attention weights for this wave's 4 column slots (head nq)
    const float al0 = attn_l[colBase + cc],      ar0 = attn_r[colBase + cc];
    const float al1 = attn_l[colBase + 16 + cc], ar1 = attn_r[colBase + 16 + cc];
    const float al2 = attn_l[colBase + 32 + cc], ar2 = attn_r[colBase + 32 + cc];
    const float al3 = attn_l[colBase + 48 + cc], ar3 = attn_r[colBase + 48 + cc];

    #pragma unroll
    for (int v = 0; v < 8; ++v) {
        const int row = rowBase + cr + v;
        float* fs = feat_src + (size_t)row * HF + colBase + cc;
        float* rs = rst      + (size_t)row * HF + colBase + cc;
        fs[ 0] = c0[v];  fs[16] = c1[v];  fs[32] = c2[v];  fs[48] = c3[v];
        rs[ 0] = d0[v] + b0;  rs[16] = d1[v] + b1;
        rs[32] = d2[v] + b2;  rs[48] = d3[v] + b3;

        // el/er partials: this wave owns all 64 f of head nq for this row
        float pl = c0[v]*al0 + c1[v]*al1 + c2[v]*al2 + c3[v]*al3;
        float pr = c0[v]*ar0 + c1[v]*ar1 + c2[v]*ar2 + c3[v]*ar3;
        #pragma unroll
        for (int m = 8; m >= 1; m >>= 1) {   // reduce across 16 lanes of this half
            pl += __shfl_xor(pl, m, 32);
            pr += __shfl_xor(pr, m, 32);
        }
        if (cc == 0) {
            el[(size_t)row * H_HEADS + nq] = pl;
            er[(size_t)row * H_HEADS + nq] = pr;
        }
    }
}

__global__ void init_attn_buffers(unsigned* __restrict__ emax_u,
                                  float* __restrict__ esum, int n) {
    int i = blockIdx.x * blockDim.x + threadIdx.x;
    if (i < n) {
        emax_u[i] = 0x007FFFFFu;   // enc(-inf)
        esum[i]   = 0.f;
    }
}

// one thread per (edge, head): logit -> a_out, atomic segment max into emax
__global__ void edge_logits_kernel(
    const int* __restrict__ src, const int* __restrict__ dst,
    const float* __restrict__ el, const float* __restrict__ er,
    float* __restrict__ a_out, unsigned* __restrict__ emax_u)
{
    long long t = (long long)blockIdx.x * blockDim.x + threadIdx.x;
    if (t >= (long long)N_EDGES * H_HEADS) return;
    int e = (int)(t >> 2), h = (int)(t & 3);
    int s = src[e], d = dst[e];
    float x = el[(size_t)s * H_HEADS + h] + er[(size_t)d * H_HEADS + h];
    x = x > 0.f ? x : NEG_SLOPE * x;
    a_out[t] = x;
    atomicMax(&emax_u[(size_t)d * H_HEADS + h], enc_f32(x));
}

// ex = exp(e - emax[dst]); atomic segment sum
__global__ void edge_exp_kernel(
    const int* __restrict__ dst, float* __restrict__ a_out,
    const unsigned* __restrict__ emax_u, float* __restrict__ esum)
{
    long long t = (long long)blockIdx.x * blockDim.x + threadIdx.x;
    if (t >= (long long)N_EDGES * H_HEADS) return;
    int e = (int)(t >> 2), h = (int)(t & 3);
    int d = dst[e];
    float m  = dec_f32(emax_u[(size_t)d * H_HEADS + h]);
    float ex = expf(a_out[t] - m);
    a_out[t] = ex;
    fadd_atomic(&esum[(size_t)d * H_HEADS + h], ex);
}

// normalize a, then SpMM: rst[dst] += a[h] * feat_src[src]; block = one edge,
// 256 threads cover (h,f). el/er/esum live in L2; feat_src row = 1KB gather.
__global__ __launch_bounds__(256) void edge_spmm_kernel(
    const int* __restrict__ src, const int* __restrict__ dst,
    const float* __restrict__ feat_src, const float* __restrict__ esum,
    float* __restrict__ a_out, float* __restrict__ rst)
{
    const int e = blockIdx.x;
    const int t = threadIdx.x;
    const int h = t >> 6;
    const int s = src[e], d = dst[e];
    float ex    = a_out[(size_t)e * H_HEADS + h];
    float denom = esum[(size_t)d * H_HEADS + h];
    float a = ex / fmaxf(denom, 1e-20f);
    __syncthreads();                         // everyone read ex before overwrite
    if ((t & 63) == 0) a_out[(size_t)e * H_HEADS + h] = a;
    float val = a * feat_src[(size_t)s * HF + t];
    fadd_atomic(&rst[(size_t)d * HF + t], val);
}

extern "C" void kernel_launch(void* const* d_in, const int* in_sizes, int n_in,
                              void* d_out, int out_size, void* d_ws, size_t ws_size,
                              hipStream_t stream) {
    const float* feat   = (const float*)d_in[0];
    const int*   src    = (const int*)  d_in[1];
    const int*   dst    = (const int*)  d_in[2];
    const float* W_fc   = (const float*)d_in[3];
    const float* attn_l = (const float*)d_in[4];
    const float* attn_r = (const float*)d_in[5];
    const float* W_res  = (const float*)d_in[6];
    const float* bias   = (const float*)d_in[7];

    float* rst   = (float*)d_out;                       // [N, H*F]
    float* a_out = rst + (size_t)N_NODES * HF;          // [E, H]

    float*    feat_src = (float*)d_ws;                  // [N, 256]
    float*    el       = feat_src + (size_t)N_NODES * HF;
    float*    er       = el + (size_t)N_NODES * H_HEADS;
    unsigned* emax_u   = (unsigned*)(er + (size_t)N_NODES * H_HEADS);
    float*    esum     = (float*)(emax_u + (size_t)N_NODES * H_HEADS);

    // 1) fused: feat_src, rst(residual+bias), el, er
    gat_fused_gemm<<<N_NODES / 32, 256, 0, stream>>>(
        feat, W_fc, W_res, bias, attn_l, attn_r, feat_src, rst, el, er);
    // 2) init segment buffers
    init_attn_buffers<<<(N_NODES * H_HEADS + 255) / 256, 256, 0, stream>>>(
        emax_u, esum, N_NODES * H_HEADS);
    // 3) edge logits + segment max
    int eth = N_EDGES * H_HEADS;
    edge_logits_kernel<<<(eth + 255) / 256, 256, 0, stream>>>(src, dst, el, er, a_out, emax_u);
    // 4) exp + segment sum
    edge_exp_kernel<<<(eth + 255) / 256, 256, 0, stream>>>(dst, a_out, emax_u, esum);
    // 5) normalize + SpMM aggregation
    edge_spmm_kernel<<<N_EDGES, 256, 0, stream>>>(src, dst, feat_src, esum, a_out, rst);
}